// Criterion_74448963109285
// MI455X (gfx1250) — compile-verified
//
#include <hip/hip_runtime.h>
#include <math.h>

typedef float v2f __attribute__((ext_vector_type(2)));
typedef float v8f __attribute__((ext_vector_type(8)));

#define NB_B 16
#define NB_E 32
#define NB_P (512 * 512)      // 262144 pixels per image
#define NB_L 3
#define TILES_PER_IMG (NB_P / 32)  // 8192 tiles of 32 pixels

// workspace layout in floats:
//  [0, 16*192)        : sums[img][row][ch], rows 0..2 = sum(emb) per label, 3..5 = sum(emb/||emb||) per label
//  [3072, 3072+48)    : counts[img][l]
//  [3120, 3120+16)    : ce partial sums per image
#define WS_CNT 3072
#define WS_CE  3120
#define WS_TOT 3136

__global__ __launch_bounds__(256) void ws_zero_kernel(float* __restrict__ ws) {
    for (int i = threadIdx.x; i < WS_TOT; i += 256) ws[i] = 0.0f;
}

// One streaming pass over embedding. Each wave: 32-pixel tiles -> LDS transpose
// -> V_WMMA_F32_16X16X4_F32 segmented reduction (rows 0..2 onehot, rows 3..5 onehot*rnorm).
// Same-wave DS ops are in-order at the LDS unit (ISA ch.11), so cross-lane
// store->load within a wave needs only a compiler fence, not a dscnt drain;
// the compiler inserts the dscnt waits required for VGPR result consumption.
__global__ __launch_bounds__(256) void seg_reduce_kernel(const float* __restrict__ emb,
                                                         const int* __restrict__ lab,
                                                         float* __restrict__ ws) {
    __shared__ float ldsB[8][NB_E * 33];  // [wave][channel*33 + pixel], pad 33 for transposed reads
    __shared__ float ldsW[8][16 * 33];    // [wave][row*33 + pixel], rows 6..15 zero

    const int img  = blockIdx.x;
    const int tid  = threadIdx.x;
    const int wid  = tid >> 5;
    const int lane = tid & 31;
    const int row  = lane & 15;
    const int half = lane >> 4;

    // zero A rows 6..15 once (read every WMMA, never rewritten)
    for (int i = lane; i < 10 * 33; i += 32) ldsW[wid][6 * 33 + i] = 0.0f;

    v8f acc0 = {};  // labels x channels 0..15
    v8f acc1 = {};  // labels x channels 16..31
    int c0 = 0, c1 = 0, c2 = 0;

    const size_t embBase   = (size_t)img * NB_E * NB_P;
    const size_t labBase   = (size_t)img * NB_P;
    const int wavesPerImg  = gridDim.y * 8;           // divides 8192 -> uniform trip counts, EXEC all-ones
    const int waveId       = blockIdx.y * 8 + wid;

    for (int t = waveId; t < TILES_PER_IMG; t += wavesPerImg) {
        const int pix = t * 32 + lane;
        const int l   = lab[labBase + pix];

        float nrm2 = 0.0f;
#pragma unroll
        for (int e = 0; e < NB_E; ++e) {             // coalesced 128B loads per channel row
            float v = emb[embBase + (size_t)e * NB_P + pix];
            ldsB[wid][e * 33 + lane] = v;
            nrm2 = fmaf(v, v, nrm2);
        }
        const float rn = 1.0f / fmaxf(sqrtf(nrm2), 1e-8f);  // CosineSimilarity eps
        const float w0 = (l == 0) ? 1.0f : 0.0f;
        const float w1 = (l == 1) ? 1.0f : 0.0f;
        const float w2 = (l == 2) ? 1.0f : 0.0f;
        c0 += (l == 0); c1 += (l == 1); c2 += (l == 2);
        ldsW[wid][0 * 33 + lane] = w0;
        ldsW[wid][1 * 33 + lane] = w1;
        ldsW[wid][2 * 33 + lane] = w2;
        ldsW[wid][3 * 33 + lane] = w0 * rn;
        ldsW[wid][4 * 33 + lane] = w1 * rn;
        ldsW[wid][5 * 33 + lane] = w2 * rn;
        asm volatile("" ::: "memory");  // compiler fence only: HW keeps same-wave DS ops in order

#pragma unroll
        for (int k = 0; k < 32; k += 4) {
            const int kk = k + 2 * half;  // VGPR0: K=k (lanes 0-15) / K=k+2 (lanes 16-31); VGPR1: +1
            v2f a, b0, b1;
            a.x  = ldsW[wid][row * 33 + kk];
            a.y  = ldsW[wid][row * 33 + kk + 1];
            b0.x = ldsB[wid][row * 33 + kk];
            b0.y = ldsB[wid][row * 33 + kk + 1];
            b1.x = ldsB[wid][(16 + row) * 33 + kk];
            b1.y = ldsB[wid][(16 + row) * 33 + kk + 1];
            acc0 = __builtin_amdgcn_wmma_f32_16x16x4_f32(false, a, false, b0, (short)0, acc0, false, false);
            acc1 = __builtin_amdgcn_wmma_f32_16x16x4_f32(false, a, false, b1, (short)0, acc1, false, false);
        }
        asm volatile("" ::: "memory");  // keep next tile's LDS stores after this tile's reads (WAR)
    }

    // C/D layout: VGPR i holds row M=i (lanes 0-15) and M=8+i (lanes 16-31), N=lane&15
    float* sums = ws + (size_t)img * 192;
#pragma unroll
    for (int i = 0; i < 8; ++i) {
        const int r = i + 8 * half;
        if (r < 6) {
            atomicAdd(&sums[r * 32 + row], acc0[i]);
            atomicAdd(&sums[r * 32 + 16 + row], acc1[i]);
        }
    }
    atomicAdd(&ws[WS_CNT + img * 3 + 0], (float)c0);
    atomicAdd(&ws[WS_CNT + img * 3 + 1], (float)c1);
    atomicAdd(&ws[WS_CNT + img * 3 + 2], (float)c2);
}

__global__ __launch_bounds__(256) void ce_kernel(const float* __restrict__ pred,
                                                 const int* __restrict__ lab,
                                                 float* __restrict__ ws) {
    const int img = blockIdx.x;
    const size_t pBase = (size_t)img * NB_L * NB_P;
    const size_t lBase = (size_t)img * NB_P;
    float acc = 0.0f;
    for (int p = blockIdx.y * 256 + threadIdx.x; p < NB_P; p += gridDim.y * 256) {
        float z0 = pred[pBase + p];
        float z1 = pred[pBase + NB_P + p];
        float z2 = pred[pBase + 2 * NB_P + p];
        int   l  = lab[lBase + p];
        float m  = fmaxf(z0, fmaxf(z1, z2));
        float s  = __expf(z0 - m) + __expf(z1 - m) + __expf(z2 - m);
        float zl = (l == 0) ? z0 : ((l == 1) ? z1 : z2);
        acc += (m + __logf(s)) - zl;
    }
    __shared__ float red[256];
    red[threadIdx.x] = acc;
    __syncthreads();
    for (int s = 128; s > 0; s >>= 1) {
        if (threadIdx.x < s) red[threadIdx.x] += red[threadIdx.x + s];
        __syncthreads();
    }
    if (threadIdx.x == 0) atomicAdd(&ws[WS_CE + img], red[0]);
}

__global__ __launch_bounds__(64) void finalize_kernel(const int* __restrict__ neighbor,
                                                      const float* __restrict__ ws,
                                                      float* __restrict__ out) {
    __shared__ float sh[NB_B];
    const int t = threadIdx.x;
    if (t < NB_B) {
        const float* sums = ws + (size_t)t * 192;
        float c[3];
        for (int l = 0; l < 3; ++l) c[l] = fmaxf(ws[WS_CNT + t * 3 + l], 1.0f);

        float nm[3][NB_E];
        for (int l = 0; l < 3; ++l) {
            float n2 = 0.0f;
            for (int e = 0; e < NB_E; ++e) {
                float m = sums[l * 32 + e] / c[l];
                nm[l][e] = m;
                n2 += m * m;
            }
            float r = 1.0f / fmaxf(sqrtf(n2), 1e-12f);  // F.normalize eps
            for (int e = 0; e < NB_E; ++e) nm[l][e] *= r;
        }
        // intra: 1 - nm[l] . sumHat[l] / count[l], labels 1..L-1, averaged
        float intra = 0.0f;
        for (int l = 1; l < 3; ++l) {
            float d = 0.0f;
            for (int e = 0; e < NB_E; ++e) d += nm[l][e] * sums[(3 + l) * 32 + e];
            intra += 1.0f - d / c[l];
        }
        intra *= 0.5f;  // / (L-1)
        // inter: neighbor-masked mean of pairwise cosines (break on first 0; dedup via bool mask)
        bool msk[3][3] = {};
        const int* nb = neighbor + t * 9;
        for (int r = 1; r < 3; ++r)
            for (int j = 0; j < 3; ++j) {
                int v = nb[r * 3 + j];
                if (v == 0) break;
                if (v >= 0 && v < 3) msk[r][v] = true;
            }
        float ssum = 0.0f; int mcnt = 0;
        for (int i = 0; i < 3; ++i)
            for (int j = 0; j < 3; ++j)
                if (msk[i][j]) {
                    float d = 0.0f;
                    for (int e = 0; e < NB_E; ++e) d += nm[i][e] * nm[j][e];
                    ssum += d; ++mcnt;
                }
        float inter = (mcnt > 0) ? ssum / (float)mcnt : 0.0f;
        float ce = ws[WS_CE + t] / (float)NB_P;
        sh[t] = intra + inter + ce;
    }
    __syncthreads();
    if (t == 0) {
        float s = 0.0f;
        for (int i = 0; i < NB_B; ++i) s += sh[i];
        out[0] = s;
    }
}

extern "C" void kernel_launch(void* const* d_in, const int* in_sizes, int n_in,
                              void* d_out, int out_size, void* d_ws, size_t ws_size,
                              hipStream_t stream) {
    const float* emb  = (const float*)d_in[0];
    const float* pred = (const float*)d_in[1];
    const int*   lab  = (const int*)d_in[2];
    const int*   nb   = (const int*)d_in[3];
    float* ws  = (float*)d_ws;
    float* out = (float*)d_out;

    ws_zero_kernel<<<1, 256, 0, stream>>>(ws);

    dim3 g1(NB_B, 64);  // 512 waves/img, 16 tiles each -> uniform, EXEC all-ones
    seg_reduce_kernel<<<g1, 256, 0, stream>>>(emb, lab, ws);

    dim3 g2(NB_B, 32);
    ce_kernel<<<g2, 256, 0, stream>>>(pred, lab, ws);

    finalize_kernel<<<1, 64, 0, stream>>>(nb, ws, out);
}